// BasicBlockInstanceNorm_85761906966882
// MI455X (gfx1250) — compile-verified
//
#include <hip/hip_runtime.h>
#include <hip/hip_bf16.h>

typedef float v2f __attribute__((ext_vector_type(2)));
typedef float v8f __attribute__((ext_vector_type(8)));

#define NPTS   100000
#define CCH    64
#define KOFF   27
#define NB     8
#define EPSI   1e-5f
#define NSLOPE 0.01f

// fp32 atomic add that should lower to global_atomic_add_f32 (HW fadd),
// not a CAS loop.
__device__ __forceinline__ void atomicAddF32(float* p, float v) {
    unsafeAtomicAdd(p, v);
}

// ---------------------------------------------------------------------------
// Zero-fill (grid-stride)
// ---------------------------------------------------------------------------
__global__ void fill_zero_kernel(float* __restrict__ p, int n) {
    for (int i = blockIdx.x * blockDim.x + threadIdx.x; i < n;
         i += gridDim.x * blockDim.x)
        p[i] = 0.0f;
}

// ---------------------------------------------------------------------------
// Sparse conv: for each offset k (blockIdx.y):
//   Y[out_map[k][i]] += X[in_map[k][i]] @ W[k]     (atomic scatter)
// Each block: 256 threads = 8 waves; each wave owns a 16-row tile.
// W[k] (64x64 f32 = 16KB) staged in LDS.
// Matrix op: V_WMMA_F32_16X16X4_F32, K-dim 64 = 16 chained WMMAs per tile,
// 4 output column tiles of 16.
// ---------------------------------------------------------------------------
__global__ __launch_bounds__(256)
void sparse_conv_kernel(const float* __restrict__ X,
                        const float* __restrict__ W,
                        const int*   __restrict__ in_map,
                        const int*   __restrict__ out_map,
                        float*       __restrict__ Y)
{
    __shared__ float Wlds[CCH * CCH];   // 16 KB

    const int k = blockIdx.y;
    const int t = threadIdx.x;

    // stage W[k] into LDS: 4096 floats = 1024 float4, 4 per thread
    {
        const float4* Wg = (const float4*)(W + k * CCH * CCH);
        float4*       Wl = (float4*)Wlds;
#pragma unroll
        for (int i = 0; i < 4; ++i)
            Wl[t + i * 256] = Wg[t + i * 256];
    }
    __syncthreads();

    const int wave    = t >> 5;
    const int lane    = t & 31;
    const int half    = lane >> 4;       // 0: lanes 0-15, 1: lanes 16-31
    const int m       = lane & 15;       // row-within-tile / col-within-tile
    const int rowbase = blockIdx.x * 128 + wave * 16;

    // gather index for this lane's A row (halves duplicate rows 0..15)
    const int ri  = rowbase + m;
    const int gin = (ri < NPTS) ? in_map[k * NPTS + ri] : 0;
    const float* Xrow = X + (long)gin * CCH;

    // A fragments: 16x4 fp32 tile per WMMA.
    // ISA layout: lanes 0-15 hold K={0,1} in VGPR{0,1}; lanes 16-31 hold K={2,3}.
    v2f a[16];
#pragma unroll
    for (int kk = 0; kk < 16; ++kk)
        a[kk] = *(const v2f*)(Xrow + kk * 4 + 2 * half);

#pragma unroll
    for (int nt = 0; nt < 4; ++nt) {
        const int n0 = nt * 16;
        v8f acc = {0.f, 0.f, 0.f, 0.f, 0.f, 0.f, 0.f, 0.f};

#pragma unroll
        for (int kk = 0; kk < 16; ++kk) {
            // B fragment 4x16: row K = vgpr + 2*half, striped across lanes
            v2f b;
            b.x = Wlds[(kk * 4 + 2 * half + 0) * CCH + n0 + m];
            b.y = Wlds[(kk * 4 + 2 * half + 1) * CCH + n0 + m];
            acc = __builtin_amdgcn_wmma_f32_16x16x4_f32(
                false, a[kk], false, b, (short)0, acc, false, false);
        }

        // scatter: C/D layout row M = r + 8*half, col N = n0 + m
#pragma unroll
        for (int r = 0; r < 8; ++r) {
            const int M    = r + 8 * half;
            const int rowM = rowbase + M;
            if (rowM < NPTS) {
                const int gout = out_map[k * NPTS + rowM];
                atomicAddF32(Y + (long)gout * CCH + n0 + m, acc[r]);
            }
        }
    }
}

// ---------------------------------------------------------------------------
// Per-(batch, channel) partial sums. batch_idx is sorted, so each thread
// accumulates in registers and flushes atomically only on batch transitions.
// Block = 256 threads: channel c = t&63, row-group g = t>>6 (4 groups x 64 rows).
// ---------------------------------------------------------------------------
__global__ __launch_bounds__(256)
void stats_kernel(const float* __restrict__ A,
                  const int*   __restrict__ bidx,
                  float* __restrict__ gsum,
                  float* __restrict__ gsq,
                  float* __restrict__ gcnt)
{
    const int t    = threadIdx.x;
    const int c    = t & 63;
    const int g    = t >> 6;
    const int row0 = blockIdx.x * 256 + g * 64;

    float s = 0.f, s2 = 0.f, cn = 0.f;
    int curb = -1;

    for (int i = 0; i < 64; ++i) {
        const int row = row0 + i;
        if (row >= NPTS) break;
        const int b = bidx[row];
        if (b != curb) {
            if (curb >= 0) {
                atomicAddF32(gsum + curb * CCH + c, s);
                atomicAddF32(gsq  + curb * CCH + c, s2);
                if (c == 0) atomicAddF32(gcnt + curb, cn);
            }
            curb = b; s = 0.f; s2 = 0.f; cn = 0.f;
        }
        const float v = A[(long)row * CCH + c];
        s += v; s2 += v * v; cn += 1.f;
    }
    if (curb >= 0) {
        atomicAddF32(gsum + curb * CCH + c, s);
        atomicAddF32(gsq  + curb * CCH + c, s2);
        if (c == 0) atomicAddF32(gcnt + curb, cn);
    }
}

// 1 block x 512 threads: mean / rstd per (b,c)
__global__ void finalize_kernel(const float* __restrict__ gsum,
                                const float* __restrict__ gsq,
                                const float* __restrict__ gcnt,
                                float* __restrict__ mean,
                                float* __restrict__ rstd)
{
    const int t  = threadIdx.x;          // 0..511
    const float cn  = gcnt[t >> 6];
    const float mu  = gsum[t] / cn;
    const float var = gsq[t] / cn - mu * mu;
    mean[t] = mu;
    rstd[t] = rsqrtf(var + EPSI);
}

// In-place: A = leaky(gamma*(A-mean)*rstd + beta)
__global__ __launch_bounds__(256)
void apply_norm_kernel(float* __restrict__ A,
                       const int*   __restrict__ bidx,
                       const float* __restrict__ mean,
                       const float* __restrict__ rstd,
                       const float* __restrict__ gamma,
                       const float* __restrict__ beta)
{
    const int idx = blockIdx.x * 256 + threadIdx.x;
    if (idx >= NPTS * CCH) return;
    const int row = idx >> 6, c = idx & 63;
    const int b   = bidx[row];
    const float x = A[idx];
    const float y = gamma[c] * (x - mean[b * CCH + c]) * rstd[b * CCH + c] + beta[c];
    A[idx] = y > 0.f ? y : NSLOPE * y;
}

// In-place on Y: Y = leaky(gamma*(Y-mean)*rstd + beta + feats)
__global__ __launch_bounds__(256)
void apply_norm_res_kernel(float* __restrict__ Y,
                           const float* __restrict__ feats,
                           const int*   __restrict__ bidx,
                           const float* __restrict__ mean,
                           const float* __restrict__ rstd,
                           const float* __restrict__ gamma,
                           const float* __restrict__ beta)
{
    const int idx = blockIdx.x * 256 + threadIdx.x;
    if (idx >= NPTS * CCH) return;
    const int row = idx >> 6, c = idx & 63;
    const int b   = bidx[row];
    const float x = Y[idx];
    float y = gamma[c] * (x - mean[b * CCH + c]) * rstd[b * CCH + c] + beta[c];
    y += feats[idx];
    Y[idx] = y > 0.f ? y : NSLOPE * y;
}

// ---------------------------------------------------------------------------
extern "C" void kernel_launch(void* const* d_in, const int* in_sizes, int n_in,
                              void* d_out, int out_size, void* d_ws, size_t ws_size,
                              hipStream_t stream)
{
    const float* feats   = (const float*)d_in[0];
    const float* W1      = (const float*)d_in[1];
    const float* g1      = (const float*)d_in[2];
    const float* b1      = (const float*)d_in[3];
    const float* W2      = (const float*)d_in[4];
    const float* g2      = (const float*)d_in[5];
    const float* b2      = (const float*)d_in[6];
    const int*   in_map  = (const int*)d_in[7];
    const int*   out_map = (const int*)d_in[8];
    const int*   bidx    = (const int*)d_in[9];
    float*       out     = (float*)d_out;

    const int NC = NPTS * CCH;

    // workspace layout (floats)
    float* A     = (float*)d_ws;        // [NC] conv1 accumulator / normed acts
    float* sbase = A + NC;
    float* sum1  = sbase;               // [512]
    float* sq1   = sbase + 512;         // [512]
    float* cnt1  = sbase + 1024;        // [64]
    float* mean1 = sbase + 1088;        // [512]
    float* rstd1 = sbase + 1600;        // [512]
    float* sum2  = sbase + 2112;        // [512]
    float* sq2   = sbase + 2624;        // [512]
    float* cnt2  = sbase + 3136;        // [64]
    float* mean2 = sbase + 3200;        // [512]
    float* rstd2 = sbase + 3712;        // [512]
    const int STATS = 4224;

    // 0) zero accumulators / stats / output
    fill_zero_kernel<<<2048, 256, 0, stream>>>(A, NC + STATS);
    fill_zero_kernel<<<2048, 256, 0, stream>>>(out, NC);

    dim3 convGrid((NPTS + 127) / 128, KOFF);
    const int statGrid  = (NPTS + 255) / 256;
    const int elemGrid  = (NC + 255) / 256;

    // 1) conv1 -> A
    sparse_conv_kernel<<<convGrid, 256, 0, stream>>>(feats, W1, in_map, out_map, A);
    // 2) instance norm 1 + leaky (in place)
    stats_kernel<<<statGrid, 256, 0, stream>>>(A, bidx, sum1, sq1, cnt1);
    finalize_kernel<<<1, 512, 0, stream>>>(sum1, sq1, cnt1, mean1, rstd1);
    apply_norm_kernel<<<elemGrid, 256, 0, stream>>>(A, bidx, mean1, rstd1, g1, b1);
    // 3) conv2 -> out
    sparse_conv_kernel<<<convGrid, 256, 0, stream>>>(A, W2, in_map, out_map, out);
    // 4) instance norm 2 + residual + leaky (in place on out)
    stats_kernel<<<statGrid, 256, 0, stream>>>(out, bidx, sum2, sq2, cnt2);
    finalize_kernel<<<1, 512, 0, stream>>>(sum2, sq2, cnt2, mean2, rstd2);
    apply_norm_res_kernel<<<elemGrid, 256, 0, stream>>>(out, feats, bidx, mean2, rstd2, g2, b2);
}